// InfluenceMatrix_15479062134803
// MI455X (gfx1250) — compile-verified
//
#include <hip/hip_runtime.h>
#include <math.h>

typedef __attribute__((ext_vector_type(2))) float v2f;
typedef __attribute__((ext_vector_type(4))) float v4f;
typedef __attribute__((ext_vector_type(8))) float v8f;

#define N_ATOMS  8192
#define FEAT     256
#define N_MOTIFS 512

// ---------------------------------------------------------------------------
// K1: row norms. One wave (32 lanes) per atom; two coalesced float4 loads per
// lane cover the 256-float row. Wave32 shfl_xor reduction. Also writes
// atomic_weights = norm / N directly to its output slot.
// ---------------------------------------------------------------------------
__global__ void norms_kernel(const float* __restrict__ g,
                             float* __restrict__ norms,
                             float* __restrict__ aw) {
    int wave = threadIdx.x >> 5;
    int lane = threadIdx.x & 31;
    int atom = blockIdx.x * 8 + wave;
    const float4* row = (const float4*)(g + (size_t)atom * FEAT);
    float4 a = row[lane];        // floats   0..127 across the wave
    float4 b = row[lane + 32];   // floats 128..255 across the wave
    float s = a.x*a.x + a.y*a.y + a.z*a.z + a.w*a.w
            + b.x*b.x + b.y*b.y + b.z*b.z + b.w*b.w;
    for (int off = 16; off; off >>= 1) s += __shfl_xor(s, off, 32);
    if (lane == 0) {
        float n = sqrtf(s);
        norms[atom] = n;
        aw[atom]    = n * (1.0f / (float)N_ATOMS);
    }
}

// ---------------------------------------------------------------------------
// K2: motif sizes -> reciprocal. One 256-thread block per motif; float4
// streaming reads, LDS tree reduction. inv[m] = 1 / (size + 1e-8).
// Also warms masks into L2 (192 MB; masks = 16 MB) for K3/K4.
// ---------------------------------------------------------------------------
__global__ void motif_size_kernel(const float* __restrict__ masks,
                                  float* __restrict__ inv) {
    __shared__ float red[256];
    int m = blockIdx.x;
    const float4* row = (const float4*)(masks + (size_t)m * N_ATOMS);
    float s = 0.0f;
    for (int k = 0; k < N_ATOMS / 4; k += 256) {        // 8 iterations
        float4 v = row[k + threadIdx.x];
        s += v.x + v.y + v.z + v.w;
    }
    red[threadIdx.x] = s;
    __syncthreads();
    for (int st = 128; st; st >>= 1) {
        if (threadIdx.x < st) red[threadIdx.x] += red[threadIdx.x + st];
        __syncthreads();
    }
    if (threadIdx.x == 0) inv[m] = 1.0f / (red[0] + 1e-8f);
}

// ---------------------------------------------------------------------------
// K3: S[i,m] = norms[i] * masks[m,i] * inv[m].  masks is [M,N] row-major but
// S is [N,M] row-major, so transpose a 32x32 tile through LDS (pad to 33 to
// stay conflict-free) so both the read and write are coalesced. S is written
// once and never re-read -> non-temporal stores keep it out of L2.
// ---------------------------------------------------------------------------
__global__ void S_kernel(const float* __restrict__ masks,
                         const float* __restrict__ norms,
                         const float* __restrict__ inv,
                         float* __restrict__ S) {
    __shared__ float tile[32][33];
    int i0 = blockIdx.x * 32;
    int m0 = blockIdx.y * 32;
    int x = threadIdx.x & 31;        // fast index
    int y = threadIdx.x >> 5;        // 0..7
    for (int j = 0; j < 4; ++j) {
        int ml = y + 8 * j;
        tile[ml][x] = masks[(size_t)(m0 + ml) * N_ATOMS + i0 + x];  // coalesced in i
    }
    __syncthreads();
    for (int j = 0; j < 4; ++j) {
        int il = y + 8 * j;
        float val = norms[i0 + il] * tile[x][il] * inv[m0 + x];
        __builtin_nontemporal_store(val, &S[(size_t)(i0 + il) * N_MOTIFS + m0 + x]);
    }
}

// ---------------------------------------------------------------------------
// K4: motif_weights = masks @ (norms/N) via V_WMMA_F32_16X16X4_F32.
// One block per 16-motif tile; wave w accumulates K-slice [w*1024, w*1024+1024).
// A[m,k] = masks[m, i+k] * norms[i+k]  (ISA 16x4 f32 A layout:
//   lanes 0-15 -> M=0..15 with K={0,1} in VGPR{0,1};
//   lanes 16-31 -> M=0..15 with K={2,3})
// B = all-ones so every column of C equals the partial motif sum (layout-proof).
// Column 0 of C/D: lane 0 holds M=0..7 in VGPRs 0..7, lane 16 holds M=8..15.
// Runs before the s_uv fill so masks is still L2-resident.
// ---------------------------------------------------------------------------
__global__ void motif_weights_wmma(const float* __restrict__ masks,
                                   const float* __restrict__ norms,
                                   float* __restrict__ mw) {
    __shared__ float partial[8][16];
    int lane = threadIdx.x & 31;
    int wave = threadIdx.x >> 5;
    int m0   = blockIdx.x * 16;
    int mrow = lane & 15;
    int kh   = (lane >> 4) * 2;                 // 0 for lanes 0-15, 2 for 16-31
    const float* rp = masks + (size_t)(m0 + mrow) * N_ATOMS;

    v8f c = {};
    v2f B; B.x = 1.0f; B.y = 1.0f;

    int kbeg = wave * (N_ATOMS / 8);
    int kend = kbeg + (N_ATOMS / 8);
    for (int i = kbeg; i < kend; i += 4) {
        int k0 = i + kh;
        v2f A;
        A.x = rp[k0]     * norms[k0];
        A.y = rp[k0 + 1] * norms[k0 + 1];
        // 8 args: (neg_a, A, neg_b, B, c_mod, C, reuse_a, reuse_b)
        c = __builtin_amdgcn_wmma_f32_16x16x4_f32(
                false, A, false, B, (short)0, c, false, false);
    }

    if (lane == 0)  { for (int v = 0; v < 8; ++v) partial[wave][v]     = c[v]; }
    if (lane == 16) { for (int v = 0; v < 8; ++v) partial[wave][v + 8] = c[v]; }
    __syncthreads();
    if (threadIdx.x < 16) {
        float s = 0.0f;
        for (int w = 0; w < 8; ++w) s += partial[w][threadIdx.x];
        mw[m0 + threadIdx.x] = s * (1.0f / (float)N_ATOMS);
    }
}

// ---------------------------------------------------------------------------
// K5: zero-fill s_uv (the 256 MB store that dominates runtime). 128-bit
// streaming stores with non-temporal hint (TH=NT) so the write-once stream
// does not sweep the 192 MB L2. Uses clang ext-vector v4f (the NT builtin
// rejects HIP's float4 class type).
// ---------------------------------------------------------------------------
__global__ void zero_suv_kernel(v4f* __restrict__ suv4) {
    size_t base = (size_t)blockIdx.x * 2048 + threadIdx.x;
    v4f z = {0.0f, 0.0f, 0.0f, 0.0f};
    for (int k = 0; k < 8; ++k)
        __builtin_nontemporal_store(z, &suv4[base + (size_t)k * 256]);
}

// ---------------------------------------------------------------------------
// K6: scatter the diagonal: s_uv[i, i] = norms[i]. Must follow the fill.
// ---------------------------------------------------------------------------
__global__ void diag_kernel(const float* __restrict__ norms,
                            float* __restrict__ suv) {
    int i = blockIdx.x * 256 + threadIdx.x;
    suv[(size_t)i * (N_ATOMS + 1)] = norms[i];
}

// ---------------------------------------------------------------------------
// Launch. d_out layout (reference return order, flat):
//   S              [8192*512] @ 0
//   atomic_weights [8192]     @ 4194304
//   motif_weights  [512]      @ 4202496
//   s_uv           [8192*8192]@ 4203008
// Workspace: norms (8192 f32) + inv motif size (512 f32) = 34 KB.
// Order: all masks readers first (L2 residency), then the NT streaming fill,
// then the diagonal scatter.
// ---------------------------------------------------------------------------
extern "C" void kernel_launch(void* const* d_in, const int* in_sizes, int n_in,
                              void* d_out, int out_size, void* d_ws, size_t ws_size,
                              hipStream_t stream) {
    const float* gnn   = (const float*)d_in[0];   // [8192, 256]
    const float* masks = (const float*)d_in[1];   // [512, 8192]

    float* out  = (float*)d_out;
    float* S    = out;
    float* aw   = out + (size_t)N_ATOMS * N_MOTIFS;
    float* mw   = aw + N_ATOMS;
    float* suv  = mw + N_MOTIFS;

    float* norms = (float*)d_ws;                  // 8192 floats
    float* inv   = norms + N_ATOMS;               // 512 floats

    norms_kernel      <<<N_ATOMS / 8, 256, 0, stream>>>(gnn, norms, aw);
    motif_size_kernel <<<N_MOTIFS,    256, 0, stream>>>(masks, inv);
    S_kernel          <<<dim3(N_ATOMS / 32, N_MOTIFS / 32), 256, 0, stream>>>(masks, norms, inv, S);
    motif_weights_wmma<<<N_MOTIFS/16, 256, 0, stream>>>(masks, norms, mw);
    zero_suv_kernel   <<<8192,        256, 0, stream>>>((v4f*)suv);
    diag_kernel       <<<N_ATOMS/256, 256, 0, stream>>>(norms, suv);
}